// EnhancedGATLayer_76476187673299
// MI455X (gfx1250) — compile-verified
//
#include <hip/hip_runtime.h>
#include <hip/hip_bf16.h>
#include <stdint.h>

#define DIMC 128
#define HEADSC 4
#define NEG_SLOPE 0.2f
#define LN_EPS 1e-5f

typedef __attribute__((ext_vector_type(2))) float v2f;
typedef __attribute__((ext_vector_type(8))) float v8f;
typedef __attribute__((ext_vector_type(4))) unsigned int u32x4;
typedef __attribute__((ext_vector_type(4))) int i32x4;
typedef __attribute__((ext_vector_type(8))) int i32x8;

// ---------- helpers ----------
__device__ __forceinline__ float lrelu(float x) { return x > 0.0f ? x : NEG_SLOPE * x; }

// monotonic float <-> orderable uint (for atomicMax-based segment max)
__device__ __forceinline__ unsigned f2k(float f) {
    unsigned b = __float_as_uint(f);
    return (b & 0x80000000u) ? ~b : (b | 0x80000000u);
}
__device__ __forceinline__ float k2f(unsigned k) {
    unsigned b = (k & 0x80000000u) ? (k & 0x7fffffffu) : ~k;
    return __uint_as_float(b);
}

// ---------- Kernel 1: h = x @ W via V_WMMA_F32_16X16X4_F32 ----------
// Block = 256 threads (8 waves). W (128x128 f32, 64KB) staged in LDS via the
// Tensor Data Mover (TENSOR_LOAD_TO_LDS), tracked with TENSORcnt.
// Wave w computes rows [blk*128 + w*16, +16) x all 128 cols (8 col tiles).
__global__ __launch_bounds__(256) void gat_gemm_wmma(
    const float* __restrict__ x, const float* __restrict__ W,
    float* __restrict__ h, int N)
{
    __shared__ float Wlds[DIMC * DIMC];  // 64 KB

    int t = threadIdx.x;

#if __has_builtin(__builtin_amdgcn_tensor_load_to_lds)
    if (t < 32) {  // single wave issues the TDM descriptor (EXEC ignored by TDM)
        uint64_t gaddr  = (uint64_t)(uintptr_t)W;
        unsigned ldsOff = (unsigned)(uintptr_t)&Wlds[0];  // addr[31:0] = LDS offset

        // D# group0: count=1 | lds_addr | global_addr[56:0] | type=2
        u32x4 g0;
        g0.x = 1u;                                   // count=1 (valid), user mode
        g0.y = ldsOff;                               // lds_addr
        g0.z = (unsigned)(gaddr & 0xffffffffu);      // global_addr[31:0]
        g0.w = ((unsigned)(gaddr >> 32) & 0x01ffffffu) | 0x80000000u;  // [56:32] | type=2

        // D# group1: data_size=4B; tensor 128x128; tile 128x128; dim0 stride 128
        i32x8 g1;
        g1[0] = 0x00020000;       // workgroup_mask=0, data_size=2 (4B)
        g1[1] = (int)(128u << 16);// tensor_dim0[15:0]=128 (atomic_barrier_addr=0)
        g1[2] = (int)(128u << 16);// tensor_dim0[31:16]=0, tensor_dim1[15:0]=128
        g1[3] = (int)(128u << 16);// tensor_dim1[31:16]=0, tile_dim0=128
        g1[4] = 128;              // tile_dim1=128, tile_dim2=0
        g1[5] = 128;              // tensor_dim0_stride[31:0]=128
        g1[6] = 0;
        g1[7] = 0;

        i32x4 gz4 = {};           // groups 2/3 unused for 2-D tile
        i32x8 gz8 = {};           // trailing group (clang-23 6-arg form)

        __builtin_amdgcn_tensor_load_to_lds(g0, g1, gz4, gz4, gz8, 0);
        __builtin_amdgcn_s_wait_tensorcnt(0);
    }
#else
    {   // fallback: cooperative 128-bit staging
        const float4* Wv = (const float4*)W;
        float4* Lv = (float4*)Wlds;
#pragma unroll
        for (int i = 0; i < 16; ++i)
            Lv[t + i * 256] = Wv[t + i * 256];
    }
#endif
    __syncthreads();

    const int wave = t >> 5;
    const int lane = t & 31;
    const int half = lane >> 4;   // 0: lanes 0-15, 1: lanes 16-31
    const int lr   = lane & 15;

    const int row0 = blockIdx.x * 128 + wave * 16;
    const int arow = row0 + lr;
    const bool rowok = arow < N;
    const float* xrow = x + (size_t)(rowok ? arow : 0) * DIMC;

    v8f acc[8] = {};  // 8 column tiles of 16x16 f32

    for (int k0 = 0; k0 < DIMC; k0 += 4) {
        // A fragment 16x4 f32 (lanes 0-15: K=k0,k0+1; lanes 16-31: K=k0+2,k0+3)
        const int k = k0 + half * 2;
        float2 av = *(const float2*)(xrow + k);
        v2f a;
        a.x = rowok ? av.x : 0.0f;
        a.y = rowok ? av.y : 0.0f;
#pragma unroll
        for (int tt = 0; tt < 8; ++tt) {
            // B fragment 4x16 f32 from LDS (mirrored layout)
            v2f b;
            b.x = Wlds[(k + 0) * DIMC + tt * 16 + lr];
            b.y = Wlds[(k + 1) * DIMC + tt * 16 + lr];
            acc[tt] = __builtin_amdgcn_wmma_f32_16x16x4_f32(
                /*neg_a=*/false, a, /*neg_b=*/false, b,
                /*c_mod=*/(short)0, acc[tt], /*reuse_a=*/false, /*reuse_b=*/false);
        }
    }

    // C/D layout: VGPR r -> row (r + half*8), lane lr -> column
#pragma unroll
    for (int tt = 0; tt < 8; ++tt) {
#pragma unroll
        for (int r = 0; r < 8; ++r) {
            int orow = row0 + r + half * 8;
            if (orow < N)
                h[(size_t)orow * DIMC + tt * 16 + lr] = acc[tt][r];
        }
    }
}

// ---------- Kernel 2: a_src/a_dst = sum(h * att) over D_HEAD ----------
__global__ __launch_bounds__(256) void gat_attn_coef(
    const float* __restrict__ h,
    const float* __restrict__ att_src, const float* __restrict__ att_dst,
    float* __restrict__ a_src, float* __restrict__ a_dst, int N)
{
    int idx = blockIdx.x * blockDim.x + threadIdx.x;  // n*HEADS + hd
    if (idx >= N * HEADSC) return;
    int hd = idx & 3;
    const float* hp = h + (size_t)(idx >> 2) * DIMC + hd * 32;
    const float* as = att_src + hd * 32;
    const float* ad = att_dst + hd * 32;
    float ssum = 0.0f, dsum = 0.0f;
#pragma unroll
    for (int d = 0; d < 32; d += 4) {
        float4 hv = *(const float4*)(hp + d);
        float4 sv = *(const float4*)(as + d);
        float4 dv = *(const float4*)(ad + d);
        ssum += hv.x * sv.x + hv.y * sv.y + hv.z * sv.z + hv.w * sv.w;
        dsum += hv.x * dv.x + hv.y * dv.y + hv.z * dv.z + hv.w * dv.w;
    }
    a_src[idx] = ssum;
    a_dst[idx] = dsum;
}

// ---------- Kernel 3: segment max via atomicMax on orderable keys ----------
__global__ __launch_bounds__(256) void gat_edge_max(
    const int* __restrict__ srcI, const int* __restrict__ dstI,
    const float* __restrict__ a_src, const float* __restrict__ a_dst,
    unsigned* __restrict__ mkey, int E, int N)
{
    int idx = blockIdx.x * blockDim.x + threadIdx.x;
    if (idx >= E + N) return;
    int s, d;
    if (idx < E) { s = srcI[idx]; d = dstI[idx]; }
    else         { s = d = idx - E; }
    float4 as = *(const float4*)(a_src + (size_t)s * 4);
    float4 ad = *(const float4*)(a_dst + (size_t)d * 4);
    unsigned* mk = mkey + (size_t)d * 4;
    atomicMax(mk + 0, f2k(lrelu(as.x + ad.x)));
    atomicMax(mk + 1, f2k(lrelu(as.y + ad.y)));
    atomicMax(mk + 2, f2k(lrelu(as.z + ad.z)));
    atomicMax(mk + 3, f2k(lrelu(as.w + ad.w)));
}

// ---------- Kernel 4: denom = segment_sum(exp(e - m)) ----------
__global__ __launch_bounds__(256) void gat_edge_sum(
    const int* __restrict__ srcI, const int* __restrict__ dstI,
    const float* __restrict__ a_src, const float* __restrict__ a_dst,
    const unsigned* __restrict__ mkey, float* __restrict__ denom, int E, int N)
{
    int idx = blockIdx.x * blockDim.x + threadIdx.x;
    if (idx >= E + N) return;
    int s, d;
    if (idx < E) { s = srcI[idx]; d = dstI[idx]; }
    else         { s = d = idx - E; }
    float4 as = *(const float4*)(a_src + (size_t)s * 4);
    float4 ad = *(const float4*)(a_dst + (size_t)d * 4);
    const unsigned* mk = mkey + (size_t)d * 4;
    float* dn = denom + (size_t)d * 4;
    atomicAdd(dn + 0, __expf(lrelu(as.x + ad.x) - k2f(mk[0])));
    atomicAdd(dn + 1, __expf(lrelu(as.y + ad.y) - k2f(mk[1])));
    atomicAdd(dn + 2, __expf(lrelu(as.z + ad.z) - k2f(mk[2])));
    atomicAdd(dn + 3, __expf(lrelu(as.w + ad.w) - k2f(mk[3])));
}

// ---------- Kernel 5: out[dst] += h[src] * alpha  (one wave32 per edge) ----------
__global__ __launch_bounds__(256) void gat_scatter(
    const int* __restrict__ srcI, const int* __restrict__ dstI,
    const float* __restrict__ h,
    const float* __restrict__ a_src, const float* __restrict__ a_dst,
    const unsigned* __restrict__ mkey, const float* __restrict__ denom,
    float* __restrict__ out, int E, int N)
{
    int gtid = blockIdx.x * 256 + threadIdx.x;
    int edge = gtid >> 5;
    int lane = gtid & 31;
    if (edge >= E + N) return;   // wave-uniform
    int s, d;
    if (edge < E) { s = srcI[edge]; d = dstI[edge]; }
    else          { s = d = edge - E; }
    const float* hp = h + (size_t)s * DIMC + lane * 4;
    __builtin_prefetch(hp, 0, 3);  // global_prefetch_b8: hide L2 gather latency
    int head = lane >> 3;  // 8 lanes (32 floats) per head
    float e = lrelu(a_src[(size_t)s * 4 + head] + a_dst[(size_t)d * 4 + head]);
    float alpha = __expf(e - k2f(mkey[(size_t)d * 4 + head])) / denom[(size_t)d * 4 + head];
    float4 hv = *(const float4*)hp;
    float* op = out + (size_t)d * DIMC + lane * 4;
    atomicAdd(op + 0, hv.x * alpha);
    atomicAdd(op + 1, hv.y * alpha);
    atomicAdd(op + 2, hv.z * alpha);
    atomicAdd(op + 3, hv.w * alpha);
}

// ---------- Kernel 6: bias + LayerNorm (one wave32 per node) ----------
__global__ __launch_bounds__(256) void gat_ln(
    float* __restrict__ out, const float* __restrict__ bias,
    const float* __restrict__ gamma, const float* __restrict__ beta, int N)
{
    int gtid = blockIdx.x * 256 + threadIdx.x;
    int node = gtid >> 5;
    int lane = gtid & 31;
    if (node >= N) return;  // wave-uniform
    float* row = out + (size_t)node * DIMC;
    float4 v  = *(const float4*)(row + lane * 4);
    float4 bv = *(const float4*)(bias + lane * 4);
    v.x += bv.x; v.y += bv.y; v.z += bv.z; v.w += bv.w;
    float s = v.x + v.y + v.z + v.w;
    float q = v.x * v.x + v.y * v.y + v.z * v.z + v.w * v.w;
#pragma unroll
    for (int off = 16; off > 0; off >>= 1) {
        s += __shfl_xor(s, off, 32);
        q += __shfl_xor(q, off, 32);
    }
    float mu  = s * (1.0f / 128.0f);
    float var = q * (1.0f / 128.0f) - mu * mu;
    float rstd = rsqrtf(var + LN_EPS);
    float4 g  = *(const float4*)(gamma + lane * 4);
    float4 b2 = *(const float4*)(beta + lane * 4);
    v.x = (v.x - mu) * rstd * g.x + b2.x;
    v.y = (v.y - mu) * rstd * g.y + b2.y;
    v.z = (v.z - mu) * rstd * g.z + b2.z;
    v.w = (v.w - mu) * rstd * g.w + b2.w;
    *(float4*)(row + lane * 4) = v;
}

extern "C" void kernel_launch(void* const* d_in, const int* in_sizes, int n_in,
                              void* d_out, int out_size, void* d_ws, size_t ws_size,
                              hipStream_t stream) {
    const float* x       = (const float*)d_in[0];
    const int*   eidx    = (const int*)d_in[1];
    const float* W       = (const float*)d_in[2];
    const float* att_src = (const float*)d_in[3];
    const float* att_dst = (const float*)d_in[4];
    const float* bias    = (const float*)d_in[5];
    const float* gamma   = (const float*)d_in[6];
    const float* beta    = (const float*)d_in[7];

    const int N = in_sizes[0] / DIMC;
    const int E = in_sizes[1] / 2;
    const int* srcI = eidx;         // edge_index[0]
    const int* dstI = eidx + E;     // edge_index[1]
    float* out = (float*)d_out;

    // workspace partition
    float*    h     = (float*)d_ws;                                   // N*128
    float*    a_src = h + (size_t)N * DIMC;                           // N*4
    float*    a_dst = a_src + (size_t)N * HEADSC;                     // N*4
    unsigned* mkey  = (unsigned*)(a_dst + (size_t)N * HEADSC);        // N*4
    float*    denom = (float*)(mkey + (size_t)N * HEADSC);            // N*4

    // zero-init accumulators (mkey==0 is below every real float key; denom==0)
    (void)hipMemsetAsync(out, 0, (size_t)N * DIMC * sizeof(float), stream);
    (void)hipMemsetAsync(mkey, 0, (size_t)N * HEADSC * sizeof(unsigned) * 2, stream);

    const int Etot = E + N;

    gat_gemm_wmma<<<(N + 127) / 128, 256, 0, stream>>>(x, W, h, N);
    gat_attn_coef<<<(N * HEADSC + 255) / 256, 256, 0, stream>>>(h, att_src, att_dst, a_src, a_dst, N);
    gat_edge_max<<<(Etot + 255) / 256, 256, 0, stream>>>(srcI, dstI, a_src, a_dst, mkey, E, N);
    gat_edge_sum<<<(Etot + 255) / 256, 256, 0, stream>>>(srcI, dstI, a_src, a_dst, mkey, denom, E, N);
    gat_scatter<<<(Etot + 7) / 8, 256, 0, stream>>>(srcI, dstI, h, a_src, a_dst, mkey, denom, out, E, N);
    gat_ln<<<(N + 7) / 8, 256, 0, stream>>>(out, bias, gamma, beta, N);
}